// BoundingBoxRegression_84370337563078
// MI455X (gfx1250) — compile-verified
//
#include <hip/hip_runtime.h>
#include <stdint.h>

// ---- problem constants (from setup_inputs) ----
#define NB   16
#define NC   64
#define NH   128
#define NW   128
#define HW   16384                 // NH*NW, one plane = 64 KB fp32
#define OUTP 192                   // 3*NC output planes per batch
#define BBOX_OFF ((size_t)NB * OUTP * HW)

typedef unsigned int v4u __attribute__((ext_vector_type(4)));
typedef int          v8i __attribute__((ext_vector_type(8)));
typedef int          v4i __attribute__((ext_vector_type(4)));

// Build the TDM descriptor (D#) groups per CDNA5 ISA.
// 2-D tile: tile_d0 elements (fp32) per row, tile_d1 rows, global row stride
// stride0 (in elements). lds_off is a byte offset in the workgroup's LDS.
__device__ __forceinline__ void tdm_desc(v4u& g0, v8i& g1,
                                         unsigned lds_off, const void* gptr,
                                         unsigned tensor_d0, unsigned tensor_d1,
                                         unsigned tile_d0, unsigned tile_d1,
                                         unsigned stride0)
{
    const uint64_t ga = (uint64_t)(uintptr_t)gptr;
    g0.x = 1u;                                             // count=1, user mode
    g0.y = lds_off;                                        // lds_addr [63:32]
    g0.z = (unsigned)(ga & 0xffffffffull);                 // global_addr lo
    g0.w = (unsigned)((ga >> 32) & 0x1ffffffull)           // global_addr[56:32]
         | (2u << 30);                                     // type=2 ("image")
    g1[0] = (int)(2u << 16);                               // data_size=2 -> 4 bytes
    g1[1] = (int)((tensor_d0 & 0xffffu) << 16);            // tensor_dim0[15:0]
    g1[2] = (int)(((tensor_d0 >> 16) & 0xffffu)            // tensor_dim0[31:16]
                | ((tensor_d1 & 0xffffu) << 16));          // tensor_dim1[15:0]
    g1[3] = (int)(((tensor_d1 >> 16) & 0xffffu)            // tensor_dim1[31:16]
                | ((tile_d0 & 0xffffu) << 16));            // tile_dim0
    g1[4] = (int)(tile_d1 & 0xffffu);                      // tile_dim1 (tile_dim2=0)
    g1[5] = (int)stride0;                                  // tensor_dim0_stride lo32
    g1[6] = 0;                                             // stride0 hi / stride1 lo
    g1[7] = 0;                                             // stride1 hi
}

__device__ __forceinline__ void tdm_load(unsigned lds_off, const void* gptr,
                                         unsigned td0, unsigned td1,
                                         unsigned t0, unsigned t1, unsigned s0)
{
    v4u g0; v8i g1; v4i z4 = {0, 0, 0, 0};
    v8i z8 = {0, 0, 0, 0, 0, 0, 0, 0};
    tdm_desc(g0, g1, lds_off, gptr, td0, td1, t0, t1, s0);
    __builtin_amdgcn_tensor_load_to_lds(g0, g1, z4, z4, z8, 0);
}

__device__ __forceinline__ void tdm_store(unsigned lds_off, const void* gptr,
                                          unsigned td0, unsigned td1,
                                          unsigned t0, unsigned t1, unsigned s0)
{
    v4u g0; v8i g1; v4i z4 = {0, 0, 0, 0};
    v8i z8 = {0, 0, 0, 0, 0, 0, 0, 0};
    tdm_desc(g0, g1, lds_off, gptr, td0, td1, t0, t1, s0);
    __builtin_amdgcn_tensor_store_from_lds(g0, g1, z4, z4, z8, 0);
}

// ---------------------------------------------------------------------------
// Kernel A: one workgroup per (b,c) plane.
//  - TDM-stage the 64KB plane into LDS (reused ~23x by the two 11-tap convs)
//  - plane max -> sigmoid threshold, score
//  - horizontal/vertical depthwise convs (weights folded: conv(x*bbx)=bbx*conv(x))
//  - write score/w/h planes to their final d_out slots (masked later in place)
//  - 5 weighted sums -> bbox row
// ---------------------------------------------------------------------------
__global__ __launch_bounds__(256)
void plane_kernel(const float* __restrict__ x,
                  const float* __restrict__ w_bbx,
                  const float* __restrict__ w_width,
                  const float* __restrict__ w_width_sh,
                  const float* __restrict__ w_height,
                  const float* __restrict__ w_height_sh,
                  float* __restrict__ out)
{
    __shared__ float xs[HW];       // 64 KB
    __shared__ float red[256];

    const int tid = threadIdx.x;
    const int p   = blockIdx.x;            // b*NC + c
    const int c   = p & (NC - 1);
    const int b   = p >> 6;

    // Async tensor DMA: whole plane -> LDS (1 row x 16384 elements)
    if (tid < 32) {
        tdm_load((unsigned)(uintptr_t)(void*)xs, x + (size_t)p * HW,
                 HW, 1, HW, 1, HW);
        __builtin_amdgcn_s_wait_tensorcnt(0);
    }
    __syncthreads();

    // plane max of x
    float m = -3.4e38f;
    for (int i = tid; i < HW; i += 256) m = fmaxf(m, xs[i]);
    red[tid] = m;
    __syncthreads();
    for (int s = 128; s > 0; s >>= 1) {
        if (tid < s) red[tid] = fmaxf(red[tid], red[tid + s]);
        __syncthreads();
    }
    const float maxx = red[0];
    __syncthreads();
    const float ps = 1.0f / (1.0f + __expf(-maxx));   // max over plane of sigmoid(x)
    const float M  = ps - 0.01f;

    // per-channel weights (uniform across block -> scalar loads)
    const float bbx = w_bbx[c];
    const float cw  = bbx * 128.0f * w_width_sh[c];   // fold bbx * W * shared 1x1
    const float ch  = bbx * 128.0f * w_height_sh[c];  // fold bbx * H * shared 1x1
    float kw[11], kh[11];
    #pragma unroll
    for (int k = 0; k < 11; ++k) {
        kw[k] = w_width[c * 11 + k];
        kh[k] = w_height[c * 11 + k];
    }

    float* outs = out + ((size_t)b * OUTP + c) * HW;  // score slot
    float* outw = outs + (size_t)NC * HW;             // width-branch slot
    float* outh = outs + (size_t)(2 * NC) * HW;       // height-branch slot

    float s0 = 0.f, s1 = 0.f, s2 = 0.f, s3 = 0.f, s4 = 0.f;
    for (int i = tid; i < HW; i += 256) {
        const int hh = i >> 7, ww = i & 127;
        const float xv  = xs[i];
        const float sig = 1.0f / (1.0f + __expf(-xv));
        const float sc  = (sig > M) ? xv : 0.0f;

        float aw = 0.f, ah = 0.f;
        #pragma unroll
        for (int k = 0; k < 11; ++k) {
            const int wc = ww - 5 + k;
            if (wc >= 0 && wc < NW) aw += kw[k] * xs[(i - ww) + wc];
            const int hc = hh - 5 + k;
            if (hc >= 0 && hc < NH) ah += kh[k] * xs[hc * NW + ww];
        }
        const float wv = aw * cw;
        const float hv = ah * ch;

        outs[i] = sc; outw[i] = wv; outh[i] = hv;

        s0 += sc;
        s1 += wv * sc;
        s2 += hv * sc;
        s3 += (float)ww * sc;
        s4 += (float)hh * sc;
    }

    float acc[5] = {s0, s1, s2, s3, s4};
    float tot[5];
    #pragma unroll
    for (int r = 0; r < 5; ++r) {
        __syncthreads();
        red[tid] = acc[r];
        __syncthreads();
        for (int s = 128; s > 0; s >>= 1) {
            if (tid < s) red[tid] += red[tid + s];
            __syncthreads();
        }
        tot[r] = red[0];
    }

    if (tid == 0) {
        const float inv = 1.0f / tot[0];              // 1/S
        const float ws = tot[1] * inv, hs = tot[2] * inv;
        const float cx = tot[3] * inv, cy = tot[4] * inv;
        const float x1 = cx - 0.5f * ws, y1 = cy - 0.5f * hs;
        float* bb = out + BBOX_OFF + (size_t)p * 6;
        bb[0] = (float)b; bb[1] = x1; bb[2] = y1;
        bb[3] = x1 + ws;  bb[4] = y1 + hs; bb[5] = ps;
    }
}

// ---------------------------------------------------------------------------
// Kernel B: one workgroup per (b,h). The 192 rows (3 families x 64 channels)
// at this (b,h) form a 2-D TDM tile (192 x 128 fp32, row stride H*W) -> 96 KB
// LDS. Each of the 384 threads owns one (family, w) column: max over 64
// channels, then keep-only-max in place. TDM store writes it back.
// ---------------------------------------------------------------------------
__global__ __launch_bounds__(384)
void mask_kernel(float* __restrict__ out)
{
    __shared__ float t[OUTP * NW];                    // 96 KB
    const int tid = threadIdx.x;
    const int b = blockIdx.x >> 7;
    const int h = blockIdx.x & 127;
    float* g = out + ((size_t)b * OUTP * NH + h) * NW;

    if (tid < 32) {
        tdm_load((unsigned)(uintptr_t)(void*)t, g, NW, OUTP, NW, OUTP, HW);
        __builtin_amdgcn_s_wait_tensorcnt(0);
    }
    __syncthreads();

    {
        const int f = tid >> 7;                       // family 0..2
        const int w = tid & 127;
        const int base = f * NC * NW + w;
        float mx = -3.4e38f;
        #pragma unroll 8
        for (int cc = 0; cc < NC; ++cc)
            mx = fmaxf(mx, t[base + cc * NW]);
        #pragma unroll 8
        for (int cc = 0; cc < NC; ++cc) {
            const float v = t[base + cc * NW];
            t[base + cc * NW] = (v == mx) ? v : 0.0f;
        }
    }
    __syncthreads();

    if (tid < 32) {
        tdm_store((unsigned)(uintptr_t)(void*)t, g, NW, OUTP, NW, OUTP, HW);
        __builtin_amdgcn_s_wait_tensorcnt(0);
    }
}

extern "C" void kernel_launch(void* const* d_in, const int* in_sizes, int n_in,
                              void* d_out, int out_size, void* d_ws, size_t ws_size,
                              hipStream_t stream)
{
    (void)in_sizes; (void)n_in; (void)d_ws; (void)ws_size; (void)out_size;
    const float* x    = (const float*)d_in[0];
    const float* wbbx = (const float*)d_in[1];
    const float* wwid = (const float*)d_in[2];
    const float* wwsh = (const float*)d_in[3];
    const float* whei = (const float*)d_in[4];
    const float* whsh = (const float*)d_in[5];
    float* out = (float*)d_out;

    plane_kernel<<<NB * NC, 256, 0, stream>>>(x, wbbx, wwid, wwsh, whei, whsh, out);
    mask_kernel<<<NB * NH, 384, 0, stream>>>(out);
}